// CubicSplineAutoregressiveSubsetTransform2d_69217692942839
// MI455X (gfx1250) — compile-verified
//
#include <hip/hip_runtime.h>

// Cubic Hermite spline forward (nsf cubic_spline, inverse=False) on gfx1250.
// Roofline: 152 B/element * 1.57M elements ~= 239 MB -> ~10.3 us at 23.3 TB/s;
// VALU ~350-400 wave-instrs / 32 elements is comparable -> keep both lean.
// No matmul structure => no WMMA (cross-lane re-layout would cost more than the
// 15 dual-issued adds a 16-wide cumsum needs). Key devices:
//   * dense clause-grouped global_load_b128 param stream (136 B/row, whole wave
//     reads a contiguous 4352 B run)
//   * LDS-backed dynamic bin gather instead of 16-deep v_cndmask chains;
//     per-thread stride of 68 floats (272 B) => 16B-aligned ds_store_b128 with
//     the optimal bank pattern (lane t starts at bank 4t mod 64: banks tiled
//     perfectly, only the mandatory 2-pass for 512 B/instr)
//   * only the two Hermite endpoint derivatives the selected bin needs

#define NBINS 16
#define MIN_BW 0.001f
#define MIN_BH 0.001f
#define SEARCH_EPSF 1e-6f

#define TPB 128
#define TSTRIDE 68            // 64 payload floats + 4 pad; 272 B => 16B-aligned slices
#define W_OFF  0
#define S_OFF  16
#define CW_OFF 32
#define CH_OFF 48

__device__ __forceinline__ float frcp(float x)  { return __builtin_amdgcn_rcpf(x); }
__device__ __forceinline__ float fexp(float x)  { return __builtin_amdgcn_exp2f(x * 1.44269504088896340736f); }
__device__ __forceinline__ float fsigmoid(float x) { return frcp(1.0f + fexp(-x)); }

__device__ __forceinline__ float dmid(float si, float sj, float wi, float wj) {
    // d_mid at the knot between adjacent bins (i, i+1)
    float m1 = fminf(fabsf(si), fabsf(sj));
    float m2 = 0.5f * (wj * si + wi * sj) * frcp(wi + wj);
    float ms = fminf(m1, m2);
    float sg = ((si > 0.0f) ? 1.0f : ((si < 0.0f) ? -1.0f : 0.0f))
             + ((sj > 0.0f) ? 1.0f : ((sj < 0.0f) ? -1.0f : 0.0f));
    return ms * sg;
}

__device__ __forceinline__ float spline_eval(float x, const float* __restrict__ tb,
                                             const float cw[17], float dlv, float drv) {
    // --- bin search (registers, fully unrolled; matches reference count-clip) ---
    int cnt = 0;
#pragma unroll
    for (int j = 0; j < 16; ++j) cnt += (x >= cw[j]) ? 1 : 0;
    cnt += (x >= (1.0f + SEARCH_EPSF)) ? 1 : 0;
    int bin = min(max(cnt - 1, 0), NBINS - 1);
    int im1 = max(bin - 1, 0);
    int ip1 = min(bin + 1, NBINS - 1);

    // --- dynamic gather from this thread's LDS slice (ds_load_b32) ---
    float w0  = tb[W_OFF  + bin];
    float wm  = tb[W_OFF  + im1];
    float wp  = tb[W_OFF  + ip1];
    float s0  = tb[S_OFF  + bin];
    float sm  = tb[S_OFF  + im1];
    float sp  = tb[S_OFF  + ip1];
    float cwb = tb[CW_OFF + bin];
    float chb = tb[CH_OFF + bin];

    // derivs[bin], derivs[bin+1]: only the two this bin needs
    float d0 = (bin == 0)         ? (dlv * 3.0f * s0) : dmid(sm, s0, wm, w0);
    float d1 = (bin == NBINS - 1) ? (drv * 3.0f * s0) : dmid(s0, sp, w0, wp);

    // Hermite cubic: ((A t + B) t + c) t + d
    float rw = frcp(w0);
    float A  = (d0 + d1 - 2.0f * s0) * rw * rw;
    float Bc = (3.0f * s0 - 2.0f * d0 - d1) * rw;
    float t  = x - cwb;
    float out = ((A * t + Bc) * t + d0) * t + chb;
    return fminf(fmaxf(out, 0.0f), 1.0f);
}

__global__ __launch_bounds__(TPB) void cubic_spline_fwd_kernel(
    const float* __restrict__ xl, const float* __restrict__ xu,
    const float* __restrict__ par,
    float* __restrict__ zl, float* __restrict__ zu, int n)
{
    __shared__ __align__(16) float smem[TPB * TSTRIDE];   // 34,816 B / block
    int i = blockIdx.x * TPB + threadIdx.x;
    if (i >= n) return;
    float* tb = &smem[threadIdx.x * TSTRIDE];             // 272 B => 16B aligned

    // --- dense param stream (8B-aligned rows; backend merges to b128 clause) ---
    const float2* p2 = (const float2*)(par + (size_t)i * 34);
    float pr[34];
#pragma unroll
    for (int j = 0; j < 17; ++j) {
        float2 v = p2[j];
        pr[2 * j] = v.x;
        pr[2 * j + 1] = v.y;
    }
    float xlo = xl[i];
    float xup = xu[i];

    // --- softmax(widths) with affine floor ---
    float w[NBINS];
    float mx = pr[0];
#pragma unroll
    for (int j = 1; j < NBINS; ++j) mx = fmaxf(mx, pr[j]);
    float sum = 0.0f;
#pragma unroll
    for (int j = 0; j < NBINS; ++j) { w[j] = fexp(pr[j] - mx); sum += w[j]; }
    float scl = (1.0f - MIN_BW * (float)NBINS) * frcp(sum);
#pragma unroll
    for (int j = 0; j < NBINS; ++j) w[j] = MIN_BW + scl * w[j];

    // --- softmax(heights) with affine floor ---
    float h[NBINS];
    mx = pr[16];
#pragma unroll
    for (int j = 1; j < NBINS; ++j) mx = fmaxf(mx, pr[16 + j]);
    sum = 0.0f;
#pragma unroll
    for (int j = 0; j < NBINS; ++j) { h[j] = fexp(pr[16 + j] - mx); sum += h[j]; }
    scl = (1.0f - MIN_BH * (float)NBINS) * frcp(sum);
#pragma unroll
    for (int j = 0; j < NBINS; ++j) h[j] = MIN_BH + scl * h[j];

    // --- cumwidths in registers (bin search uses them statically) ---
    float cw[17];
    cw[0] = 0.0f;
    float acc = 0.0f;
#pragma unroll
    for (int j = 1; j < NBINS; ++j) { acc += w[j - 1]; cw[j] = acc; }
    cw[16] = 1.0f;

    // --- stage gatherable arrays into LDS with explicit, aligned b128 stores ---
    // Same-wave DS ordering guarantees our own loads see these stores; each
    // thread only touches its own slice => no barrier needed.
    float sl[NBINS], chv[NBINS];
    acc = 0.0f;
#pragma unroll
    for (int j = 0; j < NBINS; ++j) {
        sl[j]  = h[j] * frcp(w[j]);   // slopes
        chv[j] = acc;                 // cumheights[j] = sum h[0..j-1]
        acc += h[j];
    }
#pragma unroll
    for (int q = 0; q < 4; ++q) {
        ((float4*)(tb + W_OFF))[q]  = make_float4(w[4*q],   w[4*q+1],   w[4*q+2],   w[4*q+3]);
        ((float4*)(tb + S_OFF))[q]  = make_float4(sl[4*q],  sl[4*q+1],  sl[4*q+2],  sl[4*q+3]);
        ((float4*)(tb + CW_OFF))[q] = make_float4(cw[4*q],  cw[4*q+1],  cw[4*q+2],  cw[4*q+3]);
        ((float4*)(tb + CH_OFF))[q] = make_float4(chv[4*q], chv[4*q+1], chv[4*q+2], chv[4*q+3]);
    }

    float dlv = fsigmoid(pr[32]);     // shared by both evaluations
    float drv = fsigmoid(pr[33]);

    zl[i] = spline_eval(xlo, tb, cw, dlv, drv);
    zu[i] = spline_eval(xup, tb, cw, dlv, drv);
}

extern "C" void kernel_launch(void* const* d_in, const int* in_sizes, int n_in,
                              void* d_out, int out_size, void* d_ws, size_t ws_size,
                              hipStream_t stream) {
    (void)n_in; (void)d_ws; (void)ws_size; (void)out_size;
    const float* xl  = (const float*)d_in[0];   // x_lower  [B,C,H,W]
    const float* xu  = (const float*)d_in[1];   // x_upper  [B,C,H,W]
    const float* par = (const float*)d_in[2];   // params   [B,C,H,W,34]
    float* out = (float*)d_out;                 // [z_lower | z_upper] concatenated
    int n = in_sizes[0];                        // 32*3*128*128 = 1,572,864
    int blocks = (n + TPB - 1) / TPB;
    cubic_spline_fwd_kernel<<<blocks, TPB, 0, stream>>>(xl, xu, par, out, out + n, n);
}